// AdditiveAttention_5832565588426
// MI455X (gfx1250) — compile-verified
//
#include <hip/hip_runtime.h>

typedef float v2f   __attribute__((ext_vector_type(2)));
typedef float v8f   __attribute__((ext_vector_type(8)));
typedef float f32x2 __attribute__((ext_vector_type(2)));

#if __has_builtin(__builtin_amdgcn_tanhf)
  #define FAST_TANH(x) __builtin_amdgcn_tanhf(x)
#elif __has_builtin(__builtin_amdgcn_tanh_f32)
  #define FAST_TANH(x) __builtin_amdgcn_tanh_f32(x)
#else
  #define FAST_TANH(x) tanhf(x)
#endif

// Problem constants (B=2, H=8, Q=K=512, D=64)
#define BH    16
#define QLEN  512
#define KLEN  512
#define DDIM  64
#define TQ    64
#define TK    64
#define PITCH 68

// Packed fp32 ops forced via inline asm: one main-pipe instruction per 2
// elements, balancing against v_tanh_f32 on the co-executing trans pipe.
static __device__ __forceinline__ f32x2 pk_add(f32x2 a, f32x2 b) {
  f32x2 d;
  asm("v_pk_add_f32 %0, %1, %2" : "=v"(d) : "v"(a), "v"(b));
  return d;
}
static __device__ __forceinline__ void pk_fma_acc(f32x2& acc, f32x2 a, f32x2 b) {
  asm("v_pk_fma_f32 %0, %1, %2, %0" : "+v"(acc) : "v"(a), "v"(b));
}

// One 16x16 subtile of Y = X @ W^T + bias via 16x v_wmma_f32_16x16x4_f32,
// result written to padded LDS. X = 64-row tile base in global memory.
// A-layout (16x4 f32): lane L, vgpr v -> A[L%16][(L/16)*2 + v]
// B-layout (4x16 f32): lane L, vgpr v -> B[(L/16)*2+v][L%16] = W[L%16][(L/16)*2+v]
static __device__ __forceinline__ void proj_subtile(
    const float* __restrict__ X, const float* __restrict__ W,
    const float* __restrict__ bias, float* __restrict__ lds,
    int mt, int nt, int lm, int lh)
{
  const float* arow = X + (size_t)(mt * 16 + lm) * DDIM;  // rows of X
  const float* brow = W + (size_t)(nt * 16 + lm) * DDIM;  // rows of W (= cols of W^T)

  v8f acc = {0.f, 0.f, 0.f, 0.f, 0.f, 0.f, 0.f, 0.f};
  #pragma unroll
  for (int kk = 0; kk < DDIM; kk += 4) {
    v2f a = *(const v2f*)(arow + kk + lh * 2);
    v2f b = *(const v2f*)(brow + kk + lh * 2);
    acc = __builtin_amdgcn_wmma_f32_16x16x4_f32(
        false, a, false, b, (short)0, acc, false, false);
  }

  // C/D layout: vgpr r -> M = r + 8*lh, N = lm
  const float bv    = bias[nt * 16 + lm];
  const int   mbase = mt * 16 + lh * 8;
  const int   col   = nt * 16 + lm;
  #pragma unroll
  for (int r = 0; r < 8; ++r)
    lds[(mbase + r) * PITCH + col] = acc[r] + bv;
}

// ---------------------------------------------------------------------------
// Fused additive attention:
//   out[bh,q,k] = Vb + sum_d tanh((q W^T + bw)[q,d] + (k U^T + bu)[k,d]) * Vw[d]
// Each block: project its own 64q / 64k tiles with WMMA into LDS (8 waves x
// 4 subtiles), then the trans/VALU-balanced 4x4 micro-tiled score loop.
// ---------------------------------------------------------------------------
__global__ void __launch_bounds__(256) aa_fused(
    const float* __restrict__ query, // [16,512,64]
    const float* __restrict__ keys,  // [16,512,64]
    const float* __restrict__ Wa_w, const float* __restrict__ Wa_b,
    const float* __restrict__ Ua_w, const float* __restrict__ Ua_b,
    const float* __restrict__ Vw,   const float* __restrict__ Vb,
    float* __restrict__ out)         // [16,512,512]
{
  __shared__ __align__(16) float qs[TQ * PITCH];
  __shared__ __align__(16) float ks[TK * PITCH];

  const int tid  = threadIdx.x;
  const int wave = tid >> 5;
  const int lane = tid & 31;
  const int lm   = lane & 15;
  const int lh   = lane >> 4;

  const int bh = blockIdx.z;
  const int q0 = blockIdx.x * TQ;
  const int k0 = blockIdx.y * TK;

  // ---- Phase 1: in-block projections (WMMA). 16 subtiles per tensor,
  // 8 waves -> each wave does subtiles wave*2 and wave*2+1 for q and k.
  const float* Xq = query + ((size_t)bh * QLEN + q0) * DDIM;
  const float* Xk = keys  + ((size_t)bh * KLEN + k0) * DDIM;

  #pragma unroll
  for (int t = 0; t < 2; ++t) {
    const int s  = wave * 2 + t;   // 0..15, unique per (wave,t)
    const int mt = s >> 2;
    const int nt = s & 3;
    proj_subtile(Xq, Wa_w, Wa_b, qs, mt, nt, lm, lh);
    proj_subtile(Xk, Ua_w, Ua_b, ks, mt, nt, lm, lh);
  }
  __syncthreads();

  // ---- Phase 2: score loop, 4(q) x 4(k) per-thread micro-tile.
  const int tx = tid & 15;   // k-direction (fast store dim)
  const int ty = tid >> 4;   // q-direction

  f32x2 accA[4][4];  // d%4 in {0,1}
  f32x2 accB[4][4];  // d%4 in {2,3}
  #pragma unroll
  for (int i = 0; i < 4; ++i)
    #pragma unroll
    for (int j = 0; j < 4; ++j) { accA[i][j] = (f32x2)0.f; accB[i][j] = (f32x2)0.f; }

  for (int d = 0; d < DDIM; d += 4) {
    const float4 vvv = *(const float4*)(Vw + d);   // uniform -> s_load_b128
    const f32x2 v01 = {vvv.x, vvv.y};
    const f32x2 v23 = {vvv.z, vvv.w};

    f32x2 qA[4], qB[4], kA[4], kB[4];
    #pragma unroll
    for (int i = 0; i < 4; ++i) {
      const float4 qv = *(const float4*)(qs + (ty + 16 * i) * PITCH + d);  // ds_load_b128
      qA[i] = (f32x2){qv.x, qv.y};
      qB[i] = (f32x2){qv.z, qv.w};
    }
    #pragma unroll
    for (int j = 0; j < 4; ++j) {
      const float4 kv = *(const float4*)(ks + (tx + 16 * j) * PITCH + d);  // ds_load_b128
      kA[j] = (f32x2){kv.x, kv.y};
      kB[j] = (f32x2){kv.z, kv.w};
    }

    #pragma unroll
    for (int i = 0; i < 4; ++i)
      #pragma unroll
      for (int j = 0; j < 4; ++j) {
        const f32x2 sA = pk_add(qA[i], kA[j]);                // v_pk_add_f32
        const f32x2 sB = pk_add(qB[i], kB[j]);                // v_pk_add_f32
        const f32x2 tA = {FAST_TANH(sA.x), FAST_TANH(sA.y)};  // v_tanh_f32 x2 (trans)
        const f32x2 tB = {FAST_TANH(sB.x), FAST_TANH(sB.y)};
        pk_fma_acc(accA[i][j], tA, v01);                      // v_pk_fma_f32
        pk_fma_acc(accB[i][j], tB, v23);                      // v_pk_fma_f32
      }
  }

  const float vb = Vb[0];
  #pragma unroll
  for (int i = 0; i < 4; ++i) {
    const int q = q0 + ty + 16 * i;
    float* orow = out + ((size_t)bh * QLEN + q) * KLEN + k0;
    #pragma unroll
    for (int j = 0; j < 4; ++j) {
      const f32x2 s2 = pk_add(accA[i][j], accB[i][j]);
      orow[tx + 16 * j] = (s2.x + s2.y) + vb;                 // lane-contiguous in k
    }
  }
}

// ---------------------------------------------------------------------------
extern "C" void kernel_launch(void* const* d_in, const int* in_sizes, int n_in,
                              void* d_out, int out_size, void* d_ws, size_t ws_size,
                              hipStream_t stream) {
  (void)in_sizes; (void)n_in; (void)out_size; (void)d_ws; (void)ws_size;

  const float* query = (const float*)d_in[0];  // [2,8,512,64]
  const float* keys  = (const float*)d_in[1];  // [2,8,512,64]
  const float* Wa_w  = (const float*)d_in[2];  // [64,64]
  const float* Wa_b  = (const float*)d_in[3];  // [64]
  const float* Ua_w  = (const float*)d_in[4];  // [64,64]
  const float* Ua_b  = (const float*)d_in[5];  // [64]
  const float* V_w   = (const float*)d_in[6];  // [64]
  const float* V_b   = (const float*)d_in[7];  // [1]
  float* out = (float*)d_out;                  // [2,8,512,512]

  dim3 grid(QLEN / TQ, KLEN / TK, BH);   // (8, 8, 16)
  aa_fused<<<grid, 256, 0, stream>>>(query, keys, Wa_w, Wa_b, Ua_w, Ua_b,
                                     V_w, V_b, out);
}